// SecondaryCapsuleLayer_29557964931629
// MI455X (gfx1250) — compile-verified
//
#include <hip/hip_runtime.h>

#define G 32
#define NODES 256
#define PP 8
#define IN_DIM 128
#define HALF 64
#define O 16
#define OUT_DIM 128
#define M_TOTAL (NODES * PP)      // 2048 capsule-inputs per graph
#define MTILE 64
#define MT (M_TOTAL / MTILE)      // 32 m-tiles per graph

typedef __attribute__((ext_vector_type(2))) float v2f;
typedef __attribute__((ext_vector_type(8))) float v8f;

// ---------------------------------------------------------------- utilities
__global__ __launch_bounds__(256) void kzero(float* __restrict__ p, int n) {
    int i = blockIdx.x * blockDim.x + threadIdx.x;
    if (i < n) p[i] = 0.f;
}

// xsum[g, d] = sum_m x[g, m, d]   (iteration-0 y since c is uniform 1/16)
__global__ __launch_bounds__(256) void ksum(const float* __restrict__ x,
                                            float* __restrict__ xsum) {
    int g = blockIdx.y;
    int m0 = blockIdx.x * 128;
    int t = threadIdx.x;
    int d = t & 127;
    int h = t >> 7;
    const float* xb = x + ((size_t)g * M_TOTAL + m0) * IN_DIM;
    float s = 0.f;
    for (int i = 0; i < 64; ++i)
        s += xb[(size_t)(h + 2 * i) * IN_DIM + d];
    atomicAdd(&xsum[g * IN_DIM + d], s);
}

// s_j[o,:] = scale * y[o,:] @ [Ws | Wv] + bias ; v = squash(s_j) ; optional caps out
__global__ __launch_bounds__(128) void kcaps(const float* __restrict__ ysrc,
                                             const float* __restrict__ Ws,
                                             const float* __restrict__ Wv,
                                             const float* __restrict__ bias,
                                             float* __restrict__ vout,
                                             float* __restrict__ caps,
                                             int bcast, float scale) {
    int g = blockIdx.x;
    int d = threadIdx.x;
    __shared__ float red[128];
    for (int o = 0; o < O; ++o) {
        const float* yrow = bcast ? (ysrc + (size_t)g * IN_DIM)
                                  : (ysrc + ((size_t)g * O + o) * IN_DIM);
        float s = bias[o * OUT_DIM + d];
        if (d < HALF) {
            for (int k = 0; k < HALF; ++k)
                s += scale * yrow[k] * Ws[((size_t)o * HALF + k) * HALF + d];
        } else {
            int dd = d - HALF;
            for (int k = 0; k < HALF; ++k)
                s += scale * yrow[HALF + k] * Wv[((size_t)o * HALF + k) * HALF + dd];
        }
        red[d] = s * s;
        __syncthreads();
        for (int off = 64; off > 0; off >>= 1) {
            if (d < off) red[d] += red[d + off];
            __syncthreads();
        }
        float sn = red[0];
        __syncthreads();
        float sc = sn / (1.f + sn) / (sqrtf(sn) + 1e-8f);
        float out = sc * s;
        size_t oi = ((size_t)g * O + o) * OUT_DIM + d;
        vout[oi] = out;
        if (caps) caps[oi] = out;
    }
}

// w~[g,o,d] = sum_s Ws[o,d,s] * v[g,o,s]   (s < 64: scalar part of v)
__global__ __launch_bounds__(64) void kwt(const float* __restrict__ Ws,
                                          const float* __restrict__ v,
                                          float* __restrict__ wt) {
    int o = blockIdx.x, g = blockIdx.y;
    int d = threadIdx.x;
    const float* vrow = v + ((size_t)g * O + o) * OUT_DIM;
    const float* wrow = Ws + ((size_t)o * HALF + d) * HALF;
    float s = 0.f;
    for (int k = 0; k < HALF; ++k) s += wrow[k] * vrow[k];
    wt[((size_t)g * O + o) * HALF + d] = s;
}

// Main routing iteration: agreement GEMM (WMMA) -> softmax -> c^T @ X GEMM (WMMA)
__global__ __launch_bounds__(128) void kroute(const float* __restrict__ x,
                                              const float* __restrict__ wt,
                                              float* __restrict__ blog,
                                              float* __restrict__ y,
                                              float* __restrict__ crsum,
                                              int it) {
    __shared__ float xs[MTILE * 132];  // 64x128 x-tile, pitch 132 (conflict-free)
    __shared__ float cl[MTILE * 17];   // softmax coefficients c[m, o]
    __shared__ float wts[O * 66];      // w~ tile 16x64, pitch 66

    int g = blockIdx.y;
    int m0 = blockIdx.x * MTILE;
    int t = threadIdx.x;
    int lane = t & 31;
    int w = t >> 5;
    int lo = lane & 15, hi = lane >> 4;

    // --- stage x tile (64 rows x 128 cols) into LDS, float4 coalesced
    const float4* xg = (const float4*)(x + ((size_t)g * M_TOTAL + m0) * IN_DIM);
    for (int i = 0; i < 16; ++i) {
        int lin = t + 128 * i;         // float4 index; 32 float4 per row
        int row = lin >> 5;
        int c4 = lin & 31;
        float4 val = xg[(size_t)row * 32 + c4];
        float* dst = &xs[row * 132 + c4 * 4];
        dst[0] = val.x; dst[1] = val.y; dst[2] = val.z; dst[3] = val.w;
    }
    // --- stage w~ (16x64) into LDS
    const float* wtg = wt + (size_t)g * O * HALF;
    for (int i = 0; i < 8; ++i) {
        int lin = t + 128 * i;
        int o = lin >> 6, k = lin & 63;
        wts[o * 66 + k] = wtg[lin];
    }
    __syncthreads();

    // --- Stage A: a[16m x 16o] = X_s(16x64) @ w~^T(64x16), one 16-row strip per wave
    int rb = w * 16;
    v8f acc = {0.f, 0.f, 0.f, 0.f, 0.f, 0.f, 0.f, 0.f};
    for (int k0 = 0; k0 < HALF; k0 += 4) {
        int ka = k0 + 2 * hi;
        v2f a, b;
        a.x = xs[(rb + lo) * 132 + ka];      // A[M=lo, K=ka]
        a.y = xs[(rb + lo) * 132 + ka + 1];
        b.x = wts[lo * 66 + ka];             // B[K=ka, N=o=lo] = w~[o, ka]
        b.y = wts[lo * 66 + ka + 1];
        acc = __builtin_amdgcn_wmma_f32_16x16x4_f32(false, a, false, b,
                                                    (short)0, acc, false, false);
    }
    // --- logits update + softmax over o (16-lane butterfly), c -> LDS
    for (int r = 0; r < 8; ++r) {
        int rl = r + 8 * hi;                 // local row within strip
        int m = m0 + rb + rl;
        float val = acc[r];
        if (it == 2) val += blog[((size_t)g * M_TOTAL + m) * O + lo];
        else         blog[((size_t)g * M_TOTAL + m) * O + lo] = val;
        float mx = val;
        for (int s = 1; s < 16; s <<= 1) mx = fmaxf(mx, __shfl_xor(mx, s, 32));
        float e = __expf(val - mx);
        float sum = e;
        for (int s = 1; s < 16; s <<= 1) sum += __shfl_xor(sum, s, 32);
        cl[(rb + rl) * 17 + lo] = e / sum;
    }
    __syncthreads();

    // --- Stage B: y_part[16o x 16d] = c^T(16 x 64m) @ X(64m x 16d), 2 d-tiles/wave
    for (int tile = 0; tile < 2; ++tile) {
        int d0 = w * 32 + tile * 16;
        v8f acc2 = {0.f, 0.f, 0.f, 0.f, 0.f, 0.f, 0.f, 0.f};
        for (int k0 = 0; k0 < MTILE; k0 += 4) {
            int km = k0 + 2 * hi;
            v2f a, b;
            a.x = cl[km * 17 + lo];          // A[M=o=lo, K=m=km] = c[km, lo]
            a.y = cl[(km + 1) * 17 + lo];
            b.x = xs[km * 132 + d0 + lo];    // B[K=m=km, N=d0+lo]
            b.y = xs[(km + 1) * 132 + d0 + lo];
            acc2 = __builtin_amdgcn_wmma_f32_16x16x4_f32(false, a, false, b,
                                                         (short)0, acc2, false, false);
        }
        for (int r = 0; r < 8; ++r) {
            int o = r + 8 * hi;
            atomicAdd(&y[((size_t)g * O + o) * OUT_DIM + d0 + lo], acc2[r]);
        }
    }

    // --- Stage C (final iter): crsum[g, n, o] = sum_p c[n*8+p, o]
    if (it == 2) {
        int nl = t >> 4, o = t & 15;         // 8 nodes x 16 o
        float s = 0.f;
        for (int p = 0; p < PP; ++p) s += cl[(nl * PP + p) * 17 + o];
        crsum[((size_t)g * NODES + blockIdx.x * (MTILE / PP) + nl) * O + o] = s;
    }
}

// routed[g,o,v,c] = sum_n crsum[g,n,o] * xv[g,n,v,c]
__global__ __launch_bounds__(256) void kvec(const float* __restrict__ crsum,
                                            const float* __restrict__ xv,
                                            float* __restrict__ out) {
    int g = blockIdx.x;
    int t = threadIdx.x;
    int o = t >> 4, vi = t & 15;
    float a0 = 0.f, a1 = 0.f, a2 = 0.f;
    for (int n = 0; n < NODES; ++n) {
        float cr = crsum[((size_t)g * NODES + n) * O + o];
        const float* xr = xv + (((size_t)g * NODES + n) * 16 + vi) * 3;
        a0 += cr * xr[0]; a1 += cr * xr[1]; a2 += cr * xr[2];
    }
    float* orow = out + (((size_t)g * O + o) * 16 + vi) * 3;
    orow[0] = a0; orow[1] = a1; orow[2] = a2;
}

extern "C" void kernel_launch(void* const* d_in, const int* in_sizes, int n_in,
                              void* d_out, int out_size, void* d_ws, size_t ws_size,
                              hipStream_t stream) {
    const float* x    = (const float*)d_in[0];
    const float* xv   = (const float*)d_in[1];
    const float* Ws   = (const float*)d_in[2];
    const float* Wv   = (const float*)d_in[3];
    const float* bias = (const float*)d_in[4];
    // d_in[5] = batch (unused; reference only reshapes by G)

    float* caps = (float*)d_out;                       // (G, 16, 128)
    float* vecs = (float*)d_out + (size_t)G * O * OUT_DIM;  // (G, 16, 16, 3)

    float* ws = (float*)d_ws;
    size_t off = 0;
    float* xsum = ws + off; off += (size_t)G * IN_DIM;        //   4K floats
    float* vbuf = ws + off; off += (size_t)G * O * OUT_DIM;   //  64K
    float* wtb  = ws + off; off += (size_t)G * O * HALF;      //  32K
    float* ybuf = ws + off; off += (size_t)G * O * OUT_DIM;   //  64K
    float* blog = ws + off; off += (size_t)G * M_TOTAL * O;   //   1M
    float* crs  = ws + off; off += (size_t)G * NODES * O;     // 128K

    dim3 gsum(16, G);
    dim3 grt(MT, G);
    dim3 gwt(O, G);

    // Iteration 0: c is uniform 1/16 -> s_j from plain x sum
    kzero<<<(G * IN_DIM + 255) / 256, 256, 0, stream>>>(xsum, G * IN_DIM);
    ksum<<<gsum, 256, 0, stream>>>(x, xsum);
    kcaps<<<G, 128, 0, stream>>>(xsum, Ws, Wv, bias, vbuf, nullptr, 1, 1.f / O);

    // Iteration 1
    kwt<<<gwt, 64, 0, stream>>>(Ws, vbuf, wtb);
    kzero<<<(G * O * OUT_DIM + 255) / 256, 256, 0, stream>>>(ybuf, G * O * OUT_DIM);
    kroute<<<grt, 128, 0, stream>>>(x, wtb, blog, ybuf, crs, 1);
    kcaps<<<G, 128, 0, stream>>>(ybuf, Ws, Wv, bias, vbuf, nullptr, 0, 1.f);

    // Iteration 2 (final): also emits crsum; caps written to d_out
    kwt<<<gwt, 64, 0, stream>>>(Ws, vbuf, wtb);
    kzero<<<(G * O * OUT_DIM + 255) / 256, 256, 0, stream>>>(ybuf, G * O * OUT_DIM);
    kroute<<<grt, 128, 0, stream>>>(x, wtb, blog, ybuf, crs, 2);
    kcaps<<<G, 128, 0, stream>>>(ybuf, Ws, Wv, bias, vbuf, caps, 0, 1.f);

    // Route x_vectors with final coefficients
    kvec<<<G, 256, 0, stream>>>(crs, xv, vecs);
}